// MessageAggregator_72241349919294
// MI455X (gfx1250) — compile-verified
//
#include <hip/hip_runtime.h>
#include <math.h>

#define D_DIM 64

typedef float v2f __attribute__((ext_vector_type(2)));
typedef float v8f __attribute__((ext_vector_type(8)));

__device__ __forceinline__ float celu3(float x) {
    // celu(x, alpha=3) = max(0,x) + 3*expm1(min(0,x)/3)
    return fmaxf(x, 0.0f) + 3.0f * expm1f(fminf(x, 0.0f) * (1.0f / 3.0f));
}

// order-preserving float <-> uint keys so atomicMax(uint) == float max
__device__ __forceinline__ unsigned f2key(float f) {
    unsigned u = __float_as_uint(f);
    return (u & 0x80000000u) ? ~u : (u | 0x80000000u);
}
__device__ __forceinline__ float key2f(unsigned k) {
    return __uint_as_float((k & 0x80000000u) ? (k & 0x7fffffffu) : ~k);
}

// ---- kernel 0: init accumulators -------------------------------------------
__global__ void k_init(float* __restrict__ out, unsigned* __restrict__ mkeys,
                       float* __restrict__ denom, long n_out, int nh) {
    long i = (long)blockIdx.x * blockDim.x + threadIdx.x;
    if (i < n_out) out[i] = 0.0f;
    if (i < nh) { mkeys[i] = 0u; denom[i] = 0.0f; }  // key(-large) > 0, so 0 is a safe -inf
}

// ---- kernel 1: a1 = celu(F @ W1^T) via V_WMMA_F32_16X16X4_F32 --------------
// one wave per 16-row tile of F; B = W1^T padded to 16 cols with zeros.
// B load is unconditional (clamped row) and zeroed via multiply-mask so the
// inner loop has no EXEC-diverging branches around the WMMAs.
__global__ void k_a1_wmma(const float* __restrict__ F, const float* __restrict__ W1,
                          float* __restrict__ a1, int N, int H) {
    int lane = threadIdx.x & 31;
    int wave = threadIdx.x >> 5;
    int tile = blockIdx.x * (blockDim.x >> 5) + wave;
    if (tile * 16 >= N) return;  // wave-uniform: EXEC stays all-ones for live waves

    int col   = lane & 15;               // A row index == B/N column index
    int klo   = (lane < 16) ? 0 : 2;     // K split across lane halves
    int rbase = (lane < 16) ? 0 : 8;     // C/D row base per lane half

    const float* frow  = F + (long)(tile * 16 + col) * D_DIM;
    int   bcol  = (col < H) ? col : (H - 1);          // clamped, always in-bounds
    float bmask = (col < H) ? 1.0f : 0.0f;            // zero the padded columns
    const float* wrow  = W1 + (long)bcol * D_DIM;

    v8f acc = {};
#pragma unroll
    for (int kk = 0; kk < D_DIM; kk += 4) {
        v2f a, b;
        a.x = frow[kk + klo];
        a.y = frow[kk + klo + 1];
        b.x = wrow[kk + klo] * bmask;                 // B[k][n] = W1[n][k]
        b.y = wrow[kk + klo + 1] * bmask;
        acc = __builtin_amdgcn_wmma_f32_16x16x4_f32(
            /*neg_a=*/false, a, /*neg_b=*/false, b,
            /*c_mod=*/(short)0, acc, /*reuse_a=*/false, /*reuse_b=*/false);
    }

    if (col < H) {
#pragma unroll
        for (int r = 0; r < 8; ++r) {
            int row = tile * 16 + rbase + r;
            a1[row * H + col] = celu3(acc[r]);
        }
    }
}

// ---- kernel 2: per-edge logits a = celu(a1[seg]+celu(emb)@W2^T), seg max ---
// one wave per edge; lane owns 2 of the 64 dims (coalesced float2 loads)
__global__ void k_logits(const float* __restrict__ emb, const float* __restrict__ W2,
                         const float* __restrict__ a1, const int* __restrict__ seg,
                         float* __restrict__ a_s, unsigned* __restrict__ mkeys,
                         int E, int H) {
    int lane = threadIdx.x & 31;
    int e = blockIdx.x * (blockDim.x >> 5) + (threadIdx.x >> 5);
    if (e >= E) return;

    const float2* erow = (const float2*)(emb + (long)e * D_DIM);
    float2 v = erow[lane];
    float e0 = celu3(v.x), e1 = celu3(v.y);

    float p[4];
#pragma unroll
    for (int h = 0; h < 4; ++h)
        p[h] = (h < H) ? (e0 * W2[h * D_DIM + 2 * lane] + e1 * W2[h * D_DIM + 2 * lane + 1])
                       : 0.0f;
#pragma unroll
    for (int off = 16; off > 0; off >>= 1) {
#pragma unroll
        for (int h = 0; h < 4; ++h) p[h] += __shfl_xor(p[h], off, 32);
    }

    int s = seg[e];
    if (lane < H) {
        float av = celu3(a1[s * H + lane] + p[lane]);
        a_s[(long)e * H + lane] = av;
        atomicMax(&mkeys[s * H + lane], f2key(av));
    }
}

// ---- kernel 3: ex = exp(a - m[seg]); denom += ex (in-place on a_s) ---------
__global__ void k_expsum(float* __restrict__ a_s, const unsigned* __restrict__ mkeys,
                         float* __restrict__ denom, const int* __restrict__ seg,
                         int E, int H) {
    long i = (long)blockIdx.x * blockDim.x + threadIdx.x;
    if (i >= (long)E * H) return;
    int e = (int)(i / H), h = (int)(i % H);
    int s = seg[e];
    float m = key2f(mkeys[s * H + h]);
    float exv = expf(a_s[i] - m);
    a_s[i] = exv;
    atomicAdd(&denom[s * H + h], exv);
}

// ---- kernel 4: out[seg,h,:] += (ex/denom) * celu(emb) ----------------------
// atomics land in L2 (16 MB accumulator << 192 MB L2) — no HBM amplification
__global__ void k_aggregate(const float* __restrict__ emb, const float* __restrict__ ex_s,
                            const float* __restrict__ denom, const int* __restrict__ seg,
                            float* __restrict__ out, int E, int H) {
    int lane = threadIdx.x & 31;
    int e = blockIdx.x * (blockDim.x >> 5) + (threadIdx.x >> 5);
    if (e >= E) return;
    int s = seg[e];

    const float2* erow = (const float2*)(emb + (long)e * D_DIM);
    float2 v = erow[lane];
    float e0 = celu3(v.x), e1 = celu3(v.y);

#pragma unroll
    for (int h = 0; h < 4; ++h) {
        if (h < H) {
            float att = ex_s[(long)e * H + h] / denom[s * H + h];
            float* dst = out + ((long)s * H + h) * D_DIM + 2 * lane;
            atomicAdd(dst,     att * e0);
            atomicAdd(dst + 1, att * e1);
        }
    }
}

// ---- kernel 5: out = celu(out) in place ------------------------------------
__global__ void k_celu_inplace(float* __restrict__ out, long n) {
    long i = (long)blockIdx.x * blockDim.x + threadIdx.x;
    if (i < n) out[i] = celu3(out[i]);
}

extern "C" void kernel_launch(void* const* d_in, const int* in_sizes, int n_in,
                              void* d_out, int out_size, void* d_ws, size_t ws_size,
                              hipStream_t stream) {
    const float* F   = (const float*)d_in[0];  // [N, D]
    const float* emb = (const float*)d_in[1];  // [E, D]
    const float* W1  = (const float*)d_in[2];  // [H, D]
    const float* W2  = (const float*)d_in[3];  // [H, D]
    const int*   seg = (const int*)d_in[4];    // [E]

    int N = in_sizes[0] / D_DIM;
    int E = in_sizes[1] / D_DIM;
    int H = in_sizes[2] / D_DIM;

    float* out = (float*)d_out;

    // workspace layout (floats): a_s[E*H] | mkeys[N*H] | denom[N*H] | a1[N*H]
    char* ws = (char*)d_ws;
    float*    a_s   = (float*)ws;
    unsigned* mkeys = (unsigned*)(ws + sizeof(float) * ((size_t)E * H));
    float*    denom = (float*)(ws + sizeof(float) * ((size_t)E * H + (size_t)N * H));
    float*    a1    = (float*)(ws + sizeof(float) * ((size_t)E * H + 2 * (size_t)N * H));

    long n_out = (long)N * H * D_DIM;
    int  nh    = N * H;

    k_init<<<(int)((n_out + 255) / 256), 256, 0, stream>>>(out, mkeys, denom, n_out, nh);

    {
        int tiles  = (N + 15) / 16;
        int blocks = (tiles + 7) / 8;  // 8 waves / 256-thread block
        k_a1_wmma<<<blocks, 256, 0, stream>>>(F, W1, a1, N, H);
    }

    k_logits<<<(E + 7) / 8, 256, 0, stream>>>(emb, W2, a1, seg, a_s, mkeys, E, H);

    {
        long t = (long)E * H;
        k_expsum<<<(int)((t + 255) / 256), 256, 0, stream>>>(a_s, mkeys, denom, seg, E, H);
    }

    k_aggregate<<<(E + 7) / 8, 256, 0, stream>>>(emb, a_s, denom, seg, out, E, H);

    k_celu_inplace<<<(int)((n_out + 255) / 256), 256, 0, stream>>>(out, n_out);
}